// CompatibilityLayer_36644660970122
// MI455X (gfx1250) — compile-verified
//
#include <hip/hip_runtime.h>

typedef float v2f __attribute__((ext_vector_type(2)));
typedef float v8f __attribute__((ext_vector_type(8)));

static constexpr int NN   = 8192;
static constexpr int CC   = 16;
static constexpr int BK   = 256;          // K-chunk staged in LDS
static constexpr int LSTR = BK + 4;       // LDS row stride: 260 % 64 == 4 -> conflict-free frags
static constexpr int NCH  = NN / BK;      // 32 K-chunks
static constexpr int NBLK = NN / 16;      // 512 row tiles
static constexpr int TSZ  = 16 * LSTR;    // floats per LDS tile buffer

// workspace layout (float offsets)
static constexpr size_t WS_INPUTST = 0;                 // 16 x 8192 (inputs^T)
static constexpr size_t WS_COUNTS  = (size_t)CC * NN;   // 16
static constexpr size_t WS_HPART   = WS_COUNTS + CC;    // 512 x 256 per-block H partials

__global__ void zero_kernel(float* __restrict__ counts) {
  if (threadIdx.x < CC) counts[threadIdx.x] = 0.0f;
}

// inputs = softmax(init) for unlabeled rows, one-hot(y) for labeled rows; stored transposed.
__global__ __launch_bounds__(256) void prep_kernel(
    const float* __restrict__ init_inputs, const int* __restrict__ y,
    const int* __restrict__ mask, float* __restrict__ inputsT,
    float* __restrict__ counts) {
  const int i = blockIdx.x * 256 + threadIdx.x;
  float x[CC];
  float mx = -3.402823466e38f;
#pragma unroll
  for (int c = 0; c < CC; ++c) { x[c] = init_inputs[i * CC + c]; mx = fmaxf(mx, x[c]); }
  float s = 0.0f;
#pragma unroll
  for (int c = 0; c < CC; ++c) { x[c] = expf(x[c] - mx); s += x[c]; }
  const float inv = 1.0f / s;
  const int m  = mask[i];
  const int yi = y[i];
#pragma unroll
  for (int c = 0; c < CC; ++c) {
    const float v = m ? ((c == yi) ? 1.0f : 0.0f) : x[c] * inv;
    inputsT[(size_t)c * NN + i] = v;
  }
  if (m) atomicAdd(&counts[yi], 1.0f);
}

// One block = 16 rows of A. P = A_tile @ inputs via v_wmma_f32_16x16x4_f32.
// rowsum(A) = rowsum(P) because every row of `inputs` sums to 1.
// A tiles are streamed memory->LDS with async DMA (double-buffered, NT hint).
__global__ __launch_bounds__(256) void gemm_kernel(
    const float* __restrict__ A, const float* __restrict__ inputsT,
    const int* __restrict__ y, const int* __restrict__ mask,
    float* __restrict__ Hpart) {
  __shared__ float smem[2 * TSZ];     // double-buffered A tile; reused as reduction scratch
  const int tid  = threadIdx.x;
  const int lane = tid & 31;
  const int wave = tid >> 5;
  const int ln   = lane & 15;         // A-frag row / B-frag column
  const int kk   = (lane >> 4) * 2;   // K sub-offset inside fragment (ISA 7.12.2 layout)
  const int row0 = blockIdx.x * 16;

  // each thread owns 4 float4 slots of the 16x256 tile
  const int f0 = tid * 4;
  const int r0 = f0 >> 8;             // rows r0, r0+4, r0+8, r0+12 across the 4 slots
  const int c0 = f0 & (BK - 1);

  // issue one chunk's async memory->LDS loads (4 b128 instructions per wave)
  auto stage_async = [&](int k0, int buf) {
#pragma unroll
    for (int s = 0; s < 4; ++s) {
      const int r = r0 + 4 * s;
      const float* gp = &A[(size_t)(row0 + r) * NN + (k0 + c0)];
      const unsigned lp =
          (unsigned)(size_t)&smem[buf * TSZ + r * LSTR + c0];  // LDS byte offset
      asm volatile("global_load_async_to_lds_b128 %0, %1, off th:TH_LOAD_NT"
                   :: "v"(lp), "v"(gp) : "memory");
    }
  };

  v8f acc = {};
  stage_async(0, 0);
  for (int ch = 0; ch < NCH; ++ch) {
    const int cur = ch & 1;
    if (ch + 1 < NCH) {
      stage_async((ch + 1) * BK, cur ^ 1);                 // prefetch next tile
      asm volatile("s_wait_asynccnt 0x4" ::: "memory");    // current tile's 4 are done
    } else {
      asm volatile("s_wait_asynccnt 0x0" ::: "memory");
    }
    __syncthreads();                                       // publish tile across waves
    const float* buf = &smem[cur * TSZ];
    const int k0 = ch * BK;
    const int kb = wave * 32;         // this wave's 32-wide K slice of the chunk
#pragma unroll
    for (int s = 0; s < 8; ++s) {
      const int kl = kb + s * 4;
      const v2f a = *reinterpret_cast<const v2f*>(&buf[ln * LSTR + kl + kk]);
      const v2f b = *reinterpret_cast<const v2f*>(
          &inputsT[(size_t)ln * NN + (k0 + kl + kk)]);
      acc = __builtin_amdgcn_wmma_f32_16x16x4_f32(false, a, false, b,
                                                  (short)0, acc, false, false);
    }
    __syncthreads();                  // done reading buf before it is refilled
  }

  // cross-wave reduction of the eight partial 16x16 accumulators
  float* red = smem;                  // [0,2048) per-wave C, [2048,2304) scratch, [2304,2560) Hloc
#pragma unroll
  for (int j = 0; j < 8; ++j) {
    const int m = j + 8 * (lane >> 4);        // C layout: lanes 0-15 m=j, lanes 16-31 m=j+8
    red[wave * 256 + m * 16 + ln] = acc[j];
  }
  __syncthreads();
  const int m = tid >> 4;
  float s = 0.0f;
#pragma unroll
  for (int w = 0; w < 8; ++w) s += red[w * 256 + tid];
  red[2048 + tid] = s;
  __syncthreads();
  float r = 0.0f;
#pragma unroll
  for (int c2 = 0; c2 < 16; ++c2) r += red[2048 + m * 16 + c2];   // = rowsum(A) for row m
  const float nh = s / r;             // node_h[row0+m][c]
  __syncthreads();
  red[2048 + tid] = nh;
  red[2304 + tid] = 0.0f;
  __syncthreads();
  if (tid < 16) {                     // deterministic class scatter; one thread per column
    for (int mm = 0; mm < 16; ++mm) {
      const int g = row0 + mm;
      if (mask[g]) red[2304 + y[g] * 16 + tid] += red[2048 + mm * 16 + tid];
    }
  }
  __syncthreads();
  Hpart[(size_t)blockIdx.x * 256 + tid] = red[2304 + tid];
}

// Reduce partials, divide by counts (0/0 -> NaN like the reference), NaN repair,
// then single-wave32 Sinkhorn in registers with shfl_xor butterflies.
__global__ __launch_bounds__(256) void finalize_kernel(
    const float* __restrict__ Hpart, const float* __restrict__ counts,
    float* __restrict__ out) {
  __shared__ float sH[256];
  __shared__ float miss[16];
  const int t = threadIdx.x;
  const int m = t >> 4;
  const int c = t & 15;
  float s = 0.0f;
  for (int b = 0; b < NBLK; ++b) s += Hpart[(size_t)b * 256 + t];
  const float h = s / counts[m];
  sH[t] = h;
  __syncthreads();
  const float ht = sH[c * 16 + m];    // read original transpose before overwrite
  __syncthreads();
  const float h2 = __builtin_isnan(h) ? ht : h;
  sH[t] = h2;
  __syncthreads();
  if (t < 16) {
    int cnt = 0; float rs = 0.0f;
    for (int j = 0; j < 16; ++j) {
      const float v = sH[t * 16 + j];
      if (__builtin_isnan(v)) ++cnt; else rs += v;
    }
    miss[t] = (1.0f - rs) / (float)(cnt > 1 ? cnt : 1);
  }
  __syncthreads();
  const float h3 = __builtin_isnan(h2) ? miss[m] : h2;
  sH[t] = h3;
  __syncthreads();
  if (t >= 32) return;                // only wave 0 runs the sequential loop

  const int ln = t & 15, g = t >> 4;  // lane holds column ln, rows j + 8*g
  float v[8];
#pragma unroll
  for (int j = 0; j < 8; ++j) v[j] = sH[(j + 8 * g) * 16 + ln];

  for (int it = 0; it < 3000; ++it) {
    float p = 0.0f;
#pragma unroll
    for (int j = 0; j < 8; ++j) p += v[j];
    p += __shfl_xor(p, 16, 32);       // column sums (axis=0)
    const float ic = 1.0f / p;
    float nv[8];
#pragma unroll
    for (int j = 0; j < 8; ++j) nv[j] = v[j] * ic;
#pragma unroll
    for (int j = 0; j < 8; ++j) {     // row sums (axis=1)
      float r = nv[j];
      r += __shfl_xor(r, 1, 32);
      r += __shfl_xor(r, 2, 32);
      r += __shfl_xor(r, 4, 32);
      r += __shfl_xor(r, 8, 32);
      nv[j] /= r;
    }
    float d = 0.0f;
#pragma unroll
    for (int j = 0; j < 8; ++j) { d += fabsf(nv[j] - v[j]); v[j] = nv[j]; }
    d += __shfl_xor(d, 1, 32);
    d += __shfl_xor(d, 2, 32);
    d += __shfl_xor(d, 4, 32);
    d += __shfl_xor(d, 8, 32);
    d += __shfl_xor(d, 16, 32);
    if (d < 1e-12f) break;            // matches scan early-stop semantics
  }
#pragma unroll
  for (int j = 0; j < 8; ++j) out[(j + 8 * g) * 16 + ln] = v[j];
}

extern "C" void kernel_launch(void* const* d_in, const int* in_sizes, int n_in,
                              void* d_out, int out_size, void* d_ws, size_t ws_size,
                              hipStream_t stream) {
  const float* raw_adj = (const float*)d_in[0];
  const float* init_in = (const float*)d_in[1];
  const int*   y       = (const int*)d_in[2];
  const int*   mask    = (const int*)d_in[3];
  float* ws      = (float*)d_ws;
  float* inputsT = ws + WS_INPUTST;
  float* counts  = ws + WS_COUNTS;
  float* Hpart   = ws + WS_HPART;
  float* out     = (float*)d_out;

  zero_kernel<<<1, 32, 0, stream>>>(counts);
  prep_kernel<<<NN / 256, 256, 0, stream>>>(init_in, y, mask, inputsT, counts);
  gemm_kernel<<<NBLK, 256, 0, stream>>>(raw_adj, inputsT, y, mask, Hpart);
  finalize_kernel<<<1, 256, 0, stream>>>(Hpart, counts, out);
}